// Bernstein_80693845557333
// MI455X (gfx1250) — compile-verified
//
#include <hip/hip_runtime.h>
#include <hip/hip_bf16.h>

// Problem dims
#define BB   8
#define MM   49152
#define FIN  64
#define FOUT 64
#define KK   5
#define DEG  9
#define CC   512          // FIN*BB columns per spmm matrix
#define MBLK 768          // MM / 64

typedef float v2f __attribute__((ext_vector_type(2)));
typedef float v8f __attribute__((ext_vector_type(8)));

// ---------------------------------------------------------------------------
// Phase 0: transpose [B, M, FIN] -> x0 [M, b*64 + f]
// ---------------------------------------------------------------------------
__global__ __launch_bounds__(256) void bern_transpose(const float* __restrict__ in,
                                                      float* __restrict__ x0) {
    int idx = blockIdx.x * 256 + threadIdx.x;      // M*B*16 threads
    int f4 = idx & 15;
    int b  = (idx >> 4) & 7;
    int m  = idx >> 7;
    const float4 v = *((const float4*)(in + ((size_t)b * MM + m) * FIN) + f4);
    *((float4*)(x0 + (size_t)m * CC + b * FIN) + f4) = v;
}

// ---------------------------------------------------------------------------
// Phase 1: Y = gamma * (alpha * X + beta * (S @ X)); 2 rows per 256-thread block
// ---------------------------------------------------------------------------
__global__ __launch_bounds__(256) void bern_linop(const float* __restrict__ X,
                                                  float* __restrict__ Y,
                                                  const float* __restrict__ vals,
                                                  const int* __restrict__ cols,
                                                  float alpha, float beta, float gamma) {
    int tid = threadIdx.x;
    int row = blockIdx.x * 2 + (tid >> 7);   // wave-uniform
    int c4  = tid & 127;                     // float4 column group
    int base = row * DEG;

    float4 acc = make_float4(0.f, 0.f, 0.f, 0.f);
#pragma unroll
    for (int j = 0; j < DEG; ++j) {
        int   cc = cols[base + j];
        float v  = vals[base + j];
        const float4 xv = *((const float4*)(X + (size_t)cc * CC) + c4);
        acc.x = fmaf(v, xv.x, acc.x);
        acc.y = fmaf(v, xv.y, acc.y);
        acc.z = fmaf(v, xv.z, acc.z);
        acc.w = fmaf(v, xv.w, acc.w);
    }

    float4 r;
    if (alpha != 0.0f) {
        const float4 xr = *((const float4*)(X + (size_t)row * CC) + c4);
        r.x = gamma * fmaf(alpha, xr.x, beta * acc.x);
        r.y = gamma * fmaf(alpha, xr.y, beta * acc.y);
        r.z = gamma * fmaf(alpha, xr.z, beta * acc.z);
        r.w = gamma * fmaf(alpha, xr.w, beta * acc.w);
    } else {
        float gb = gamma * beta;
        r.x = gb * acc.x;  r.y = gb * acc.y;  r.z = gb * acc.z;  r.w = gb * acc.w;
    }
    *((float4*)(Y + (size_t)row * CC) + c4) = r;
}

// ---------------------------------------------------------------------------
// Phase 2: out[b*M+m, fo] (+)= sum_f Y[m, b*64+f] * Weff[f, fo]
//   Weff[f,fo] = kernel[(f*6+i)*64+fo]  (+ theta5*kernel[(f*6+5)*64+fo] if i==4,
//                folding the stale-x3 term stack[5] = theta5*stack[4]).
// Block = 128 threads (4 waves). Each wave: one 16x64 strip of one batch b.
// 16 chained V_WMMA_F32_16X16X4_F32 per 16x16 tile (K = 64).
//
// Weights are staged into LDS PRE-SWIZZLED into exact B-fragment order so each
// B fragment pair is one conflict-free ds_load_2addr_b64, and the C strip is
// addressed off a single per-lane base pointer so loads/stores use immediate
// offsets (no per-element 64-bit address arithmetic).
// ---------------------------------------------------------------------------
__global__ __launch_bounds__(128) void bern_gemm_acc(const float* __restrict__ Y,
                                                     const float* __restrict__ kern,
                                                     float* __restrict__ out,
                                                     int ii, int initOut) {
    __shared__ v2f Wsw[16 * 4 * 32];       // 2048 float2 = 16 KB

    int tid = threadIdx.x;
    const float th5 = 1.0f / 32.0f;
    // Stage pre-swizzled effective weights (each thread: 16 entries, 32 loads).
    for (int e = tid; e < 16 * 4 * 32; e += 128) {
        int ln  = e & 31;
        int g   = e >> 5;                          // g = kk*4 + t  (0..63)
        int row = (g >> 2) * 4 + ((ln >> 4) << 1); // 4*kk + 2*half
        int col = (g & 3) * 16 + (ln & 15);        // t*16 + l15
        float w0 = kern[(size_t)(row * 6 + ii) * FOUT + col];
        float w1 = kern[(size_t)((row + 1) * 6 + ii) * FOUT + col];
        if (ii == 4) {
            w0 = fmaf(th5, kern[(size_t)(row * 6 + 5) * FOUT + col], w0);
            w1 = fmaf(th5, kern[(size_t)((row + 1) * 6 + 5) * FOUT + col], w1);
        }
        v2f p; p.x = w0; p.y = w1;
        Wsw[e] = p;
    }
    __syncthreads();

    int lane  = tid & 31;
    int wv    = tid >> 5;
    int b     = blockIdx.x / MBLK;
    int mblk  = blockIdx.x % MBLK;
    int m0    = mblk * 64 + wv * 16;

    int lhalf = lane >> 4;                 // 0: lanes 0-15, 1: lanes 16-31
    int l15   = lane & 15;

    // A-fragment row pointer: row m0+l15, cols b*64 + k, half-wave K offset
    const float* Arow = Y + (size_t)(m0 + l15) * CC + b * FIN + lhalf * 2;
    const v2f*   Wp   = Wsw + lane;

    // Single per-lane base for the 16x64 C strip; element (t, r) at
    // constant offset r*FOUT + t*16  (max 1984 B -> imm offsets).
    float* op = out + ((size_t)b * MM + m0 + lhalf * 8) * FOUT + l15;

    v8f c[4];
#pragma unroll
    for (int t = 0; t < 4; ++t)
#pragma unroll
        for (int r = 0; r < 8; ++r) c[t][r] = 0.0f;

    if (!initOut) {
#pragma unroll
        for (int t = 0; t < 4; ++t)
#pragma unroll
            for (int r = 0; r < 8; ++r)
                c[t][r] = op[r * FOUT + t * 16];
    }

#pragma unroll
    for (int kkk = 0; kkk < 16; ++kkk) {
        // A fragment: lanes 0-15 -> K = 4kk,4kk+1 ; lanes 16-31 -> K = 4kk+2,4kk+3
        v2f a = *(const v2f*)(Arow + kkk * 4);
#pragma unroll
        for (int t = 0; t < 4; ++t) {
            v2f bb = Wp[(kkk * 4 + t) * 32];   // ds_load, imm offset
            c[t] = __builtin_amdgcn_wmma_f32_16x16x4_f32(
                /*neg_a=*/false, a, /*neg_b=*/false, bb,
                /*c_mod=*/(short)0, c[t], /*reuse_a=*/false, /*reuse_b=*/false);
        }
    }

#pragma unroll
    for (int t = 0; t < 4; ++t)
#pragma unroll
        for (int r = 0; r < 8; ++r)
            op[r * FOUT + t * 16] = c[t][r];
}

// ---------------------------------------------------------------------------
// Host orchestration
// ---------------------------------------------------------------------------
extern "C" void kernel_launch(void* const* d_in, const int* in_sizes, int n_in,
                              void* d_out, int out_size, void* d_ws, size_t ws_size,
                              hipStream_t stream) {
    const float* input  = (const float*)d_in[0];   // [B, M, FIN]
    const float* L_vals = (const float*)d_in[1];   // [NNZ]
    const float* kernel = (const float*)d_in[2];   // [(K+1)*FIN, FOUT]
    // d_in[3] = L_rows: fixed-degree sorted rows => row(n) = n/DEG, unused
    const int*   L_cols = (const int*)d_in[4];     // [NNZ]
    float* out = (float*)d_out;

    const size_t MSZ = (size_t)MM * CC;            // floats per matrix
    float* bufs[3] = { (float*)d_ws,
                       (float*)d_ws + MSZ,
                       (float*)d_ws + 2 * MSZ };
    (void)in_sizes; (void)n_in; (void)out_size; (void)ws_size;

    const float theta[6] = { 1.f/32.f, 5.f/32.f, 10.f/32.f, 10.f/32.f, 5.f/32.f, 1.f/32.f };

    // x0 -> bufs[0]
    int zi = 0;
    bern_transpose<<<(MM * BB * 16) / 256, 256, 0, stream>>>(input, bufs[0]);

    const int linop_grid = MM / 2;
    const int gemm_grid  = BB * MBLK;

    for (int i = 0; i < 5; ++i) {
        int p0 = (zi + 1) % 3, p1 = (zi + 2) % 3;
        const float* src = bufs[zi];               // S^i x0
        int dsti = p0;
        int nA = 5 - i;
        for (int j = 0; j < nA; ++j) {
            float gamma = (j == nA - 1) ? theta[i] : 1.0f;
            bern_linop<<<linop_grid, 256, 0, stream>>>(src, bufs[dsti], L_vals, L_cols,
                                                       2.0f, -1.0f, gamma);
            src  = bufs[dsti];
            dsti = (dsti == p0) ? p1 : p0;
        }
        // src now holds stack[i] = theta_i * A^{5-i} S^i x0 ; bufs[dsti] is free
        bern_gemm_acc<<<gemm_grid, 128, 0, stream>>>(src, kernel, out, i, (i == 0) ? 1 : 0);

        if (i < 4) {
            // z_{i+1} = S z_i into the free temp (S^5 x0 is dead code in the ref)
            bern_linop<<<linop_grid, 256, 0, stream>>>(bufs[zi], bufs[dsti], L_vals, L_cols,
                                                       0.0f, 1.0f, 1.0f);
            zi = dsti;
        }
    }
}